// S4Layer_49813030699449
// MI455X (gfx1250) — compile-verified
//
#include <hip/hip_runtime.h>

// ---------------------------------------------------------------------------
// S4 layer for MI455X (gfx1250, wave32, WMMA, async-to-LDS).
//   U = X @ B^T                      (WMMA f32 16x16x4 GEMM, async LDS staging)
//   chunked linear scan (K=32, 64 chunks):
//     P2: local scans, 1 WG per chunk (LDS state, no grid sync)
//     Pb: boundary recurrence with A^32 (5 WMMA squarings), 64 grid barriers
//     P3: carry propagation, 1 WG per chunk, accumulate in place
//   Y = S @ C^T + D*x                (WMMA GEMM) -> d_out
//   LayerNorm in place on d_out
// ---------------------------------------------------------------------------

typedef float v2f __attribute__((ext_vector_type(2)));
typedef float v8f __attribute__((ext_vector_type(8)));
typedef int   v4i __attribute__((__vector_size__(16)));

#define DM    1024
#define SEQ   2048
#define BATCH 4
#define ROWS  (BATCH * SEQ)     // 8192
#define CHUNK 32
#define NCHUNK (SEQ / CHUNK)    // 64

#if __has_builtin(__builtin_amdgcn_global_load_async_to_lds_b128) && \
    __has_builtin(__builtin_amdgcn_s_wait_asynccnt)
#define HAVE_ASYNC_LDS 1
#else
#define HAVE_ASYNC_LDS 0
#endif

typedef __attribute__((address_space(1))) v4i* gbl_v4i_p;   // global int4*
typedef __attribute__((address_space(3))) v4i* lds_v4i_p;   // LDS int4*

// ------------------------- WMMA NT GEMM ------------------------------------
// Out[r][n] = sum_k In[r][k] * W[n][k]  (+ D[0]*Add[r][n] if Add != null)
// Block tile 64(M) x 128(N), K staged 32 at a time into double-buffered LDS
// via GLOBAL_LOAD_ASYNC_TO_LDS_B128 (ASYNCcnt), 8 waves (2x4), each wave a
// 32x32 tile = 4 x v_wmma_f32_16x16x4_f32 accumulators.
#define XS_STRIDE 36
#define WS_STRIDE 36

__device__ __forceinline__ void copy16_async(const float* g, float* l) {
#if HAVE_ASYNC_LDS
  __builtin_amdgcn_global_load_async_to_lds_b128((gbl_v4i_p)g, (lds_v4i_p)l, 0, 0);
#else
  *(float4*)l = *(const float4*)g;
#endif
}

// 6 async ops per thread per stage (2 X-chunks + 4 W-chunks).
__device__ __forceinline__ void stage_tile(const float* __restrict__ In,
                                           const float* __restrict__ W,
                                           int Kdim, int rb, int cb, int kb,
                                           float* Xs, float* Ws, int tid) {
#pragma unroll
  for (int p0 = 0; p0 < 512; p0 += 256) {
    const int p = p0 + tid;
    const int r = p >> 3, c4 = (p & 7) << 2;
    copy16_async(&In[(size_t)(rb + r) * Kdim + kb + c4], &Xs[r * XS_STRIDE + c4]);
  }
#pragma unroll
  for (int p0 = 0; p0 < 1024; p0 += 256) {
    const int p = p0 + tid;
    const int r = p >> 3, c4 = (p & 7) << 2;
    copy16_async(&W[(size_t)(cb + r) * Kdim + kb + c4], &Ws[r * WS_STRIDE + c4]);
  }
}

__launch_bounds__(256)
__global__ void gemm_nt_wmma(const float* __restrict__ In,
                             const float* __restrict__ W,
                             const float* __restrict__ Add,
                             const float* __restrict__ Dp,
                             float* __restrict__ Out, int Kdim) {
  __shared__ float Xs[2][64 * XS_STRIDE];
  __shared__ float Ws[2][128 * WS_STRIDE];
  const int tid  = threadIdx.x;
  const int lane = tid & 31;
  const int wave = tid >> 5;
  const int wm   = wave & 1;            // 0..1  (M sub-block of 32)
  const int wn   = wave >> 1;           // 0..3  (N sub-block of 32)
  const int rb   = blockIdx.x * 64;
  const int cb   = blockIdx.y * 128;
  const int lm   = lane & 15;
  const int lk   = (lane >> 4) * 2;     // K pair select per ISA A/B layout

  v8f acc[2][2] = {};

  const int nstage = Kdim / 32;
  stage_tile(In, W, Kdim, rb, cb, 0, Xs[0], Ws[0], tid);

  for (int s = 0; s < nstage; ++s) {
    const int cur = s & 1;
    if (s + 1 < nstage) {
      stage_tile(In, W, Kdim, rb, cb, (s + 1) * 32, Xs[cur ^ 1], Ws[cur ^ 1], tid);
#if HAVE_ASYNC_LDS
      __builtin_amdgcn_s_wait_asynccnt(6);   // retire stage s (in-order)
#endif
    } else {
#if HAVE_ASYNC_LDS
      __builtin_amdgcn_s_wait_asynccnt(0);
#endif
    }
    __syncthreads();                         // stage s visible to all waves
    const float* xs = Xs[cur];
    const float* ws = Ws[cur];
#pragma unroll
    for (int kk = 0; kk < 32; kk += 4) {
      v2f a[2], b[2];
#pragma unroll
      for (int mi = 0; mi < 2; ++mi)
        a[mi] = *(const v2f*)&xs[(wm * 32 + mi * 16 + lm) * XS_STRIDE + kk + lk];
#pragma unroll
      for (int ni = 0; ni < 2; ++ni)
        b[ni] = *(const v2f*)&ws[(wn * 32 + ni * 16 + lm) * WS_STRIDE + kk + lk];
#pragma unroll
      for (int mi = 0; mi < 2; ++mi)
#pragma unroll
        for (int ni = 0; ni < 2; ++ni)
          acc[mi][ni] = __builtin_amdgcn_wmma_f32_16x16x4_f32(
              false, a[mi], false, b[ni], (short)0, acc[mi][ni], false, false);
    }
    __syncthreads();                         // done reading before overwrite
  }

  const float Dv = Add ? Dp[0] : 0.0f;
#pragma unroll
  for (int mi = 0; mi < 2; ++mi) {
#pragma unroll
    for (int ni = 0; ni < 2; ++ni) {
#pragma unroll
      for (int vg = 0; vg < 8; ++vg) {
        const int m = rb + wm * 32 + mi * 16 + (lane >> 4) * 8 + vg;
        const int n = cb + wn * 32 + ni * 16 + lm;
        float v = acc[mi][ni][vg];
        if (Add) v += Dv * Add[(size_t)m * DM + n];
        Out[(size_t)m * DM + n] = v;
      }
    }
  }
}

// ------------------------- 1024x1024 transpose -----------------------------
__launch_bounds__(256)
__global__ void transpose1024(const float* __restrict__ Pin, float* __restrict__ Pout) {
  __shared__ float t[32][33];
  const int bx = blockIdx.x * 32, by = blockIdx.y * 32;
  const int x = threadIdx.x & 31, yb = threadIdx.x >> 5;
#pragma unroll
  for (int i = 0; i < 32; i += 8)
    t[yb + i][x] = Pin[(size_t)(by + yb + i) * DM + bx + x];
  __syncthreads();
#pragma unroll
  for (int i = 0; i < 32; i += 8)
    Pout[(size_t)(bx + yb + i) * DM + by + x] = t[x][yb + i];
}

// ------------------------- local chunk scan (P2) ---------------------------
// One WG per chunk. State kept in LDS as s[i][b] (float4 over batch).
// 8 lanes cooperate per output row j; A streamed from L2 each step with
// prefetch of the next row group.
__launch_bounds__(256)
__global__ void scan_local(float* __restrict__ U, const float* __restrict__ A,
                           float* __restrict__ Lend) {
  __shared__ float sbuf[2][DM][4];     // [phase][i][b] : 2 x 16 KB
  const int c    = blockIdx.x;
  const int tid  = threadIdx.x;
  const int p    = tid & 7;            // partial-sum lane within row group
  const int jrow = tid >> 3;           // 0..31

  for (int q = tid; q < DM * 4; q += 256) sbuf[0][q >> 2][q & 3] = 0.0f;
  __syncthreads();

  int cur = 0;
  for (int tt = 0; tt < CHUNK; ++tt) {
    const int t = c * CHUNK + tt;
    for (int jb = 0; jb < 32; ++jb) {
      const int j = jb * 32 + jrow;
      const float* __restrict__ Arow = &A[(size_t)j * DM];
      if (jb + 1 < 32) __builtin_prefetch(&A[(size_t)(j + 32) * DM + p * 4], 0, 1);
      float y0 = 0.f, y1 = 0.f, y2 = 0.f, y3 = 0.f;
      for (int ii = 0; ii < 32; ++ii) {
        const int i0 = ii * 32 + p * 4;
        const float4 a4 = *(const float4*)&Arow[i0];
        const float4 s0 = *(const float4*)&sbuf[cur][i0 + 0][0];
        const float4 s1 = *(const float4*)&sbuf[cur][i0 + 1][0];
        const float4 s2 = *(const float4*)&sbuf[cur][i0 + 2][0];
        const float4 s3 = *(const float4*)&sbuf[cur][i0 + 3][0];
        y0 += a4.x * s0.x + a4.y * s1.x + a4.z * s2.x + a4.w * s3.x;
        y1 += a4.x * s0.y + a4.y * s1.y + a4.z * s2.y + a4.w * s3.y;
        y2 += a4.x * s0.z + a4.y * s1.z + a4.z * s2.z + a4.w * s3.z;
        y3 += a4.x * s0.w + a4.y * s1.w + a4.z * s2.w + a4.w * s3.w;
      }
#pragma unroll
      for (int d = 1; d < 8; d <<= 1) {
        y0 += __shfl_xor(y0, d, 32);
        y1 += __shfl_xor(y1, d, 32);
        y2 += __shfl_xor(y2, d, 32);
        y3 += __shfl_xor(y3, d, 32);
      }
      if (p == 0) {
        float y[4] = {y0, y1, y2, y3};
#pragma unroll
        for (int b = 0; b < 4; ++b) {
          const size_t idx = ((size_t)b * SEQ + t) * DM + j;
          const float v = y[b] + U[idx];
          U[idx] = v;                       // L_t overwrites u_t in place
          sbuf[cur ^ 1][j][b] = v;
          if (tt == CHUNK - 1)
            Lend[(size_t)c * 4096 + (size_t)b * DM + j] = v;
        }
      }
    }
    __syncthreads();
    cur ^= 1;
  }
}

// ------------------------- carry scan (P3) ---------------------------------
// carry_{t} = carry_{t-1} @ A^T starting from Sc[c-1]; U[t] += carry_t.
__launch_bounds__(256)
__global__ void scan_carry(float* __restrict__ U, const float* __restrict__ A,
                           const float* __restrict__ Sc) {
  __shared__ float sbuf[2][DM][4];
  const int c = blockIdx.x;
  if (c == 0) return;                   // carry into chunk 0 is zero
  const int tid  = threadIdx.x;
  const int p    = tid & 7;
  const int jrow = tid >> 3;

  for (int q = tid; q < DM * 4; q += 256)
    sbuf[0][q >> 2][q & 3] = Sc[(size_t)(c - 1) * 4096 + (size_t)(q & 3) * DM + (q >> 2)];
  __syncthreads();

  int cur = 0;
  for (int tt = 0; tt < CHUNK; ++tt) {
    const int t = c * CHUNK + tt;
    for (int jb = 0; jb < 32; ++jb) {
      const int j = jb * 32 + jrow;
      const float* __restrict__ Arow = &A[(size_t)j * DM];
      if (jb + 1 < 32) __builtin_prefetch(&A[(size_t)(j + 32) * DM + p * 4], 0, 1);
      float y0 = 0.f, y1 = 0.f, y2 = 0.f, y3 = 0.f;
      for (int ii = 0; ii < 32; ++ii) {
        const int i0 = ii * 32 + p * 4;
        const float4 a4 = *(const float4*)&Arow[i0];
        const float4 s0 = *(const float4*)&sbuf[cur][i0 + 0][0];
        const float4 s1 = *(const float4*)&sbuf[cur][i0 + 1][0];
        const float4 s2 = *(const float4*)&sbuf[cur][i0 + 2][0];
        const float4 s3 = *(const float4*)&sbuf[cur][i0 + 3][0];
        y0 += a4.x * s0.x + a4.y * s1.x + a4.z * s2.x + a4.w * s3.x;
        y1 += a4.x * s0.y + a4.y * s1.y + a4.z * s2.y + a4.w * s3.y;
        y2 += a4.x * s0.z + a4.y * s1.z + a4.z * s2.z + a4.w * s3.z;
        y3 += a4.x * s0.w + a4.y * s1.w + a4.z * s2.w + a4.w * s3.w;
      }
#pragma unroll
      for (int d = 1; d < 8; d <<= 1) {
        y0 += __shfl_xor(y0, d, 32);
        y1 += __shfl_xor(y1, d, 32);
        y2 += __shfl_xor(y2, d, 32);
        y3 += __shfl_xor(y3, d, 32);
      }
      if (p == 0) {
        float y[4] = {y0, y1, y2, y3};
#pragma unroll
        for (int b = 0; b < 4; ++b) {
          const size_t idx = ((size_t)b * SEQ + t) * DM + j;
          U[idx] += y[b];                 // s_t = L_t + carry_t
          sbuf[cur ^ 1][j][b] = y[b];
        }
      }
    }
    __syncthreads();
    cur ^= 1;
  }
}

// ------------------------- boundary scan (Pb) ------------------------------
__device__ inline void grid_bar(unsigned* bar, unsigned e, int numWG) {
  __syncthreads();
  if (threadIdx.x == 0) {
    __threadfence();
    __hip_atomic_fetch_add(bar, 1u, __ATOMIC_ACQ_REL, __HIP_MEMORY_SCOPE_AGENT);
    const unsigned target = (e + 1u) * (unsigned)numWG;
    while (__hip_atomic_load(bar, __ATOMIC_ACQUIRE, __HIP_MEMORY_SCOPE_AGENT) < target)
      __builtin_amdgcn_s_sleep(2);
    __threadfence();
  }
  __syncthreads();
}

// S_c = Lend[c] + S_{c-1} @ (A^32)^T ; 64 WGs each own 16 output columns,
// their 64 KB slice of A^32 cached in LDS for all 63 sequential steps.
__launch_bounds__(256)
__global__ void boundary_scan(const float* __restrict__ Lend,
                              const float* __restrict__ M32,
                              float* __restrict__ Sc,
                              unsigned* __restrict__ bar) {
  __shared__ float Ms[16 * DM];   // 64 KB slice
  __shared__ float Sl[4 * DM];    // 16 KB staged previous boundary state
  const int tid = threadIdx.x;
  const int jb  = blockIdx.x * 16;

  for (int q = tid; q < 4096; q += 256) {
    const int r = q >> 8, c4 = (q & 255) << 2;
    *(float4*)&Ms[r * DM + c4] = *(const float4*)&M32[(size_t)(jb + r) * DM + c4];
  }
  // Sc[0] = Lend[0]
  if (tid < 64) {
    const int b = tid >> 4, jj = tid & 15;
    Sc[(size_t)b * DM + jb + jj] = Lend[(size_t)b * DM + jb + jj];
  }
  grid_bar(bar, 0, NCHUNK);

  for (int c = 1; c < NCHUNK; ++c) {
    for (int q = tid; q < 1024; q += 256)
      ((float4*)Sl)[q] = ((const float4*)&Sc[(size_t)(c - 1) * 4096])[q];
    __syncthreads();
    const int o = tid >> 2, part = tid & 3;
    const int jj = o >> 2, b = o & 3;
    float sum = 0.f;
    for (int i = part * 256; i < part * 256 + 256; i += 4) {
      const float4 s4 = *(const float4*)&Sl[b * DM + i];
      const float4 m4 = *(const float4*)&Ms[jj * DM + i];
      sum += s4.x * m4.x + s4.y * m4.y + s4.z * m4.z + s4.w * m4.w;
    }
    sum += __shfl_xor(sum, 1, 32);
    sum += __shfl_xor(sum, 2, 32);
    if (part == 0)
      Sc[(size_t)c * 4096 + (size_t)b * DM + jb + jj] =
          Lend[(size_t)c * 4096 + (size_t)b * DM + jb + jj] + sum;
    grid_bar(bar, (unsigned)c, NCHUNK);
  }
}

// ------------------------- LayerNorm (in place) ----------------------------
__launch_bounds__(256)
__global__ void layernorm_rows(float* __restrict__ Y,
                               const float* __restrict__ g,
                               const float* __restrict__ bta) {
  __shared__ float rS[8], rQ[8], mv[2];
  const int tid = threadIdx.x;
  float* yr = Y + (size_t)blockIdx.x * DM;
  float4 v = *(float4*)&yr[tid * 4];
  float s = v.x + v.y + v.z + v.w;
  float q = v.x * v.x + v.y * v.y + v.z * v.z + v.w * v.w;
#pragma unroll
  for (int d = 1; d < 32; d <<= 1) {
    s += __shfl_xor(s, d, 32);
    q += __shfl_xor(q, d, 32);
  }
  if ((tid & 31) == 0) { rS[tid >> 5] = s; rQ[tid >> 5] = q; }
  __syncthreads();
  if (tid == 0) {
    float S = 0.f, Q = 0.f;
    for (int w = 0; w < 8; ++w) { S += rS[w]; Q += rQ[w]; }
    const float mu  = S * (1.0f / DM);
    const float var = Q * (1.0f / DM) - mu * mu;
    mv[0] = mu;
    mv[1] = rsqrtf(var + 1e-5f);
  }
  __syncthreads();
  const float mu = mv[0], r = mv[1];
  const float4 g4 = *(const float4*)&g[tid * 4];
  const float4 b4 = *(const float4*)&bta[tid * 4];
  v.x = (v.x - mu) * r * g4.x + b4.x;
  v.y = (v.y - mu) * r * g4.y + b4.y;
  v.z = (v.z - mu) * r * g4.z + b4.z;
  v.w = (v.w - mu) * r * g4.w + b4.w;
  *(float4*)&yr[tid * 4] = v;
}

// ------------------------- host launcher -----------------------------------
extern "C" void kernel_launch(void* const* d_in, const int* in_sizes, int n_in,
                              void* d_out, int out_size, void* d_ws, size_t ws_size,
                              hipStream_t stream) {
  const float* x  = (const float*)d_in[0];
  const float* A  = (const float*)d_in[1];
  const float* B  = (const float*)d_in[2];
  const float* Cw = (const float*)d_in[3];
  const float* Dp = (const float*)d_in[4];
  const float* g  = (const float*)d_in[5];
  const float* bt = (const float*)d_in[6];
  float* out = (float*)d_out;

  char* ws = (char*)d_ws;
  float*    U    = (float*)(ws);                             // 32 MB: u_t -> L_t -> s_t
  float*    Pa   = (float*)(ws + (size_t)32 * 1024 * 1024);  // 4 MB
  float*    Pb   = (float*)(ws + (size_t)36 * 1024 * 1024);  // 4 MB
  float*    Pt   = (float*)(ws + (size_t)40 * 1024 * 1024);  // 4 MB
  float*    Lend = (float*)(ws + (size_t)44 * 1024 * 1024);  // 1 MB
  float*    Sc   = (float*)(ws + (size_t)45 * 1024 * 1024);  // 1 MB
  unsigned* bar  = (unsigned*)(ws + (size_t)46 * 1024 * 1024);

  (void)hipMemsetAsync(bar, 0, 256, stream);

  // U = X @ B^T
  gemm_nt_wmma<<<dim3(ROWS / 64, DM / 128), 256, 0, stream>>>(x, B, nullptr, nullptr, U, DM);

  // A^32 by repeated squaring (NT gemm needs transpose of right operand)
  transpose1024<<<dim3(32, 32), 256, 0, stream>>>(A, Pt);
  gemm_nt_wmma<<<dim3(DM / 64, DM / 128), 256, 0, stream>>>(A, Pt, nullptr, nullptr, Pa, DM); // A^2
  float* cur = Pa;
  float* nxt = Pb;
  for (int it = 0; it < 4; ++it) {      // -> A^4, A^8, A^16, A^32
    transpose1024<<<dim3(32, 32), 256, 0, stream>>>(cur, Pt);
    gemm_nt_wmma<<<dim3(DM / 64, DM / 128), 256, 0, stream>>>(cur, Pt, nullptr, nullptr, nxt, DM);
    float* tmp = cur; cur = nxt; nxt = tmp;
  }

  // chunked scan
  scan_local<<<NCHUNK, 256, 0, stream>>>(U, A, Lend);
  boundary_scan<<<NCHUNK, 256, 0, stream>>>(Lend, cur, Sc, bar);
  scan_carry<<<NCHUNK, 256, 0, stream>>>(U, A, Sc);

  // Y = S @ C^T + D*x  -> d_out, then LayerNorm in place
  gemm_nt_wmma<<<dim3(ROWS / 64, DM / 128), 256, 0, stream>>>(U, Cw, x, Dp, out, DM);
  layernorm_rows<<<ROWS, 256, 0, stream>>>(out, g, bt);
}